// HyperTensorGraphConvolution_15255723835411
// MI455X (gfx1250) — compile-verified
//
#include <hip/hip_runtime.h>

// ---------------------------------------------------------------------------
// HyperTensorGraphConvolution for MI455X (gfx1250, wave32)
//   N=50000 nodes, E=16384 edges, K=12, S=8, D=128
//   out = rownorm(scatter_add(clip(H), sqrt(mean_s clip(H)^2))) @ W + bias
// ---------------------------------------------------------------------------

#define D 128
#define KM1_INV (1.0f / 11.0f)   // 1/(K-1)
#define MIN_V 1e-7f
#define MAX_V 10.0f

typedef __attribute__((ext_vector_type(2))) float v2f;
typedef __attribute__((ext_vector_type(8))) float v8f;

// Async global->LDS staging (CDNA5 GLOBAL_LOAD_ASYNC_TO_LDS_B128, ASYNCcnt).
#if defined(__has_builtin)
#  if __has_builtin(__builtin_amdgcn_global_load_async_to_lds_b128) && \
      __has_builtin(__builtin_amdgcn_s_wait_asynccnt)
#    define USE_ASYNC_LDS 1
#  endif
#endif
#ifndef USE_ASYNC_LDS
#  define USE_ASYNC_LDS 0
#endif

// ---------------------------------------------------------------------------
// Kernel 1: Hc = clip(H), Hp = Hc^2, sig = Hc   (float4 streaming)
// ---------------------------------------------------------------------------
__global__ __launch_bounds__(256) void prep_kernel(const float* __restrict__ H,
                                                   float* __restrict__ Hp,
                                                   float* __restrict__ sig,
                                                   int total4) {
    int i = blockIdx.x * blockDim.x + threadIdx.x;
    if (i >= total4) return;
    float4 h = reinterpret_cast<const float4*>(H)[i];
    float4 hc;
    hc.x = fminf(fmaxf(h.x, MIN_V), MAX_V);
    hc.y = fminf(fmaxf(h.y, MIN_V), MAX_V);
    hc.z = fminf(fmaxf(h.z, MIN_V), MAX_V);
    hc.w = fminf(fmaxf(h.w, MIN_V), MAX_V);
    float4 hp = make_float4(hc.x * hc.x, hc.y * hc.y, hc.z * hc.z, hc.w * hc.w);
    reinterpret_cast<float4*>(sig)[i] = hc;
    reinterpret_cast<float4*>(Hp)[i]  = hp;
}

// ---------------------------------------------------------------------------
// Kernel 2: per (edge,member) row: contrib = sqrt(sum_s Hp[nbr]/11),
//           atomic-add into sig[edge_nodes[row]].
//           128 threads handle one row (one feature each), 2 rows per block.
// ---------------------------------------------------------------------------
__global__ __launch_bounds__(256) void edge_scatter_kernel(const float* __restrict__ Hp,
                                                           const int* __restrict__ edge_nodes,
                                                           const int* __restrict__ nbr_ids,
                                                           float* __restrict__ sig,
                                                           int EK) {
    int row = blockIdx.x * 2 + (threadIdx.x >> 7);   // (e,k) flat index
    if (row >= EK) return;
    int d = threadIdx.x & (D - 1);
    const int* nb = nbr_ids + (size_t)row * 8;
    float sum = 0.0f;
#pragma unroll
    for (int s = 0; s < 8; ++s) {
        int n = nb[s];                               // uniform across the 128-group
        sum += Hp[(size_t)n * D + d];                // coalesced 512B gather (L2-resident)
    }
    float contrib = __builtin_sqrtf(sum * KM1_INV);
    int node = edge_nodes[row];
    __hip_atomic_fetch_add(&sig[(size_t)node * D + d], contrib,
                           __ATOMIC_RELAXED, __HIP_MEMORY_SCOPE_AGENT);
}

// ---------------------------------------------------------------------------
// Kernel 3: in-place row normalize. One wave32 per node row; each lane owns a
//           float4 (32 lanes * 4 = 128), wave reduction via shfl_xor.
// ---------------------------------------------------------------------------
__global__ __launch_bounds__(256) void rownorm_kernel(float* __restrict__ sig, int Nrows) {
    int wave = (blockIdx.x * blockDim.x + threadIdx.x) >> 5;
    int lane = threadIdx.x & 31;
    if (wave >= Nrows) return;
    float4* rowp = reinterpret_cast<float4*>(sig + (size_t)wave * D);
    float4 v = rowp[lane];
    float s = v.x + v.y + v.z + v.w;
#pragma unroll
    for (int off = 16; off > 0; off >>= 1) s += __shfl_xor(s, off, 32);
    float rinv = (s > 0.0f) ? (1.0f / s) : 0.0f;
    v.x *= rinv; v.y *= rinv; v.z *= rinv; v.w *= rinv;
    rowp[lane] = v;
}

// ---------------------------------------------------------------------------
// Kernel 4: out[N,128] = sig[N,128] @ W[128,128] + bias, via
//           V_WMMA_F32_16X16X4_F32. W staged in LDS (64 KB of the 320 KB WGP
//           LDS) using GLOBAL_LOAD_ASYNC_TO_LDS_B128 when available. 8
//           waves/block; each wave computes a 16x128 output stripe:
//           32 k-steps x 8 n-tiles = 256 WMMA issues.
//
// Fragment layouts (CDNA5 ISA 7.12.2, f32 operands):
//   A 16x4 : lanes 0-15 -> {K=0 (v0), K=1 (v1)}, lanes 16-31 -> {K=2, K=3};
//            row M = lane & 15 in both halves.
//   B 4x16 : v0 -> rows {0 | 2} across lane halves, v1 -> rows {1 | 3};
//            col N = lane & 15.
//   C/D    : vgpr v -> row v (lanes 0-15) / row v+8 (lanes 16-31), col = lane&15.
// ---------------------------------------------------------------------------
__global__ __launch_bounds__(256) void gemm_wmma_kernel(const float* __restrict__ AH,
                                                        const float* __restrict__ W,
                                                        const float* __restrict__ bias,
                                                        float* __restrict__ out,
                                                        int Nrows) {
    __shared__ float Wlds[D * D];                    // 64 KB

    const int tid = threadIdx.x;
#if USE_ASYNC_LDS
    {
        // 16 passes x 256 threads x 16 B = 64 KB, direct global->LDS DMA path.
        // Builtin expects non-const int4-vector pointers in AS1 / AS3.
        typedef int v4i_vs __attribute__((vector_size(16)));
        typedef __attribute__((address_space(1))) v4i_vs* gvec_p;
        typedef __attribute__((address_space(3))) v4i_vs* lvec_p;
        gvec_p g = (gvec_p)W;
        lvec_p l = (lvec_p)Wlds;
#pragma unroll
        for (int i = 0; i < 16; ++i) {
            int idx = i * 256 + tid;                 // 16-byte units
            __builtin_amdgcn_global_load_async_to_lds_b128(g + idx, l + idx, 0, 0);
        }
        __builtin_amdgcn_s_wait_asynccnt(0);         // LDS visible before barrier
    }
#else
    // Fallback: cooperative load, 4096 float4s across 256 threads (16 each).
    for (int i = tid; i < (D * D) / 4; i += 256) {
        reinterpret_cast<float4*>(Wlds)[i] = reinterpret_cast<const float4*>(W)[i];
    }
#endif
    __syncthreads();

    const int wave = tid >> 5;
    const int lane = tid & 31;
    const int half = lane >> 4;                      // 0: lanes 0-15, 1: lanes 16-31
    const int m    = lane & 15;
    const int row_base = blockIdx.x * 128 + wave * 16;
    if (row_base >= Nrows) return;                   // wave-uniform exit (after barrier)

    // Bias folded into accumulator init: every row of column n gets bias[n].
    v8f acc[8];
#pragma unroll
    for (int nt = 0; nt < 8; ++nt) {
        float bv = bias[nt * 16 + m];
#pragma unroll
        for (int v = 0; v < 8; ++v) acc[nt][v] = bv;
    }

    // A row per lane (clamped for the ragged tail; duplicate reads are benign,
    // stores are predicated below so EXEC stays all-ones around the WMMAs).
    int rowA = row_base + m;
    if (rowA >= Nrows) rowA = Nrows - 1;
    const float2* A2 = reinterpret_cast<const float2*>(AH + (size_t)rowA * D);

#pragma unroll 4
    for (int k = 0; k < D; k += 4) {
        float2 af = A2[(k >> 1) + half];             // floats [k+2*half, k+2*half+1]
        v2f a; a.x = af.x; a.y = af.y;
        const int kr0 = (k + 2 * half) * D;          // B v0 row
        const int kr1 = kr0 + D;                     // B v1 row
#pragma unroll
        for (int nt = 0; nt < 8; ++nt) {
            const int n = nt * 16 + m;
            v2f b; b.x = Wlds[kr0 + n]; b.y = Wlds[kr1 + n];
            acc[nt] = __builtin_amdgcn_wmma_f32_16x16x4_f32(
                /*neg_a=*/false, a, /*neg_b=*/false, b,
                /*c_mod=*/(short)0, acc[nt],
                /*reuse_a=*/false, /*reuse_b=*/false);
        }
    }

    // Store D: vgpr v holds row (v + 8*half), column nt*16 + m.
#pragma unroll
    for (int nt = 0; nt < 8; ++nt) {
#pragma unroll
        for (int v = 0; v < 8; ++v) {
            int r = row_base + v + 8 * half;
            if (r < Nrows) out[(size_t)r * D + nt * 16 + m] = acc[nt][v];
        }
    }
}

// ---------------------------------------------------------------------------
// Host launcher. ws layout: [Hp : N*128 f32][sig : N*128 f32]  (~51.2 MB)
// ---------------------------------------------------------------------------
extern "C" void kernel_launch(void* const* d_in, const int* in_sizes, int n_in,
                              void* d_out, int out_size, void* d_ws, size_t ws_size,
                              hipStream_t stream) {
    const float* H          = (const float*)d_in[0];
    const int*   edge_nodes = (const int*)d_in[1];
    const int*   nbr_ids    = (const int*)d_in[2];
    const float* W          = (const float*)d_in[3];
    const float* bias       = (const float*)d_in[4];
    float*       out        = (float*)d_out;

    const int Nrows = in_sizes[0] / D;               // 50000
    const int EK    = in_sizes[1];                   // E*K = 196608

    float* Hp  = (float*)d_ws;
    float* sig = Hp + (size_t)Nrows * D;

    const int total4 = (Nrows * D) / 4;
    prep_kernel<<<(total4 + 255) / 256, 256, 0, stream>>>(H, Hp, sig, total4);

    edge_scatter_kernel<<<(EK + 1) / 2, 256, 0, stream>>>(Hp, edge_nodes, nbr_ids, sig, EK);

    const int waves = Nrows;                         // one wave32 per row
    rownorm_kernel<<<(waves * 32 + 255) / 256, 256, 0, stream>>>(sig, Nrows);

    gemm_wmma_kernel<<<(Nrows + 127) / 128, 256, 0, stream>>>(sig, W, bias, out, Nrows);
}